// LightGCN_67714454389280
// MI455X (gfx1250) — compile-verified
//
#include <hip/hip_runtime.h>
#include <math.h>

#define NUM_LAYERS 3
#define D 128           // feature dim (fixed by reference)
#define D4 (D / 4)      // 32 float4 per row == one wave32's lanes

#define WAVES_PER_BLOCK 8
#define SPMM_BLOCK (WAVES_PER_BLOCK * 32)
#define SCAN_B 1024
#define PF 4            // async pipeline depth (LDS buffers per wave)

// --- gfx1250 async global->LDS copy (ASYNCcnt path), with safe fallbacks ----
#if defined(__HIP_DEVICE_COMPILE__) && defined(__gfx1250__) && \
    __has_builtin(__builtin_amdgcn_global_load_async_to_lds_b128)
#define HAVE_ASYNC_LDS 1
typedef int v4i_t __attribute__((ext_vector_type(4)));
typedef __attribute__((address_space(1))) v4i_t* gvec_p;   // global int4*
typedef __attribute__((address_space(3))) v4i_t* lvec_p;   // LDS int4*
#else
#define HAVE_ASYNC_LDS 0
#endif

#if defined(__HIP_DEVICE_COMPILE__) && defined(__gfx1250__) && \
    __has_builtin(__builtin_amdgcn_s_wait_asynccnt)
#define WAIT_ASYNC(n) __builtin_amdgcn_s_wait_asynccnt(n)
#elif defined(__HIP_DEVICE_COMPILE__) && defined(__gfx1250__)
#define WAIT_ASYNC(n) asm volatile("s_wait_asynccnt %0" ::"n"(n) : "memory")
#else
#define WAIT_ASYNC(n)
#endif

// ---------------------------------------------------------------------------
// K0: x0 = concat(user_w, item_w); acc(d_out) = same.  float4-vectorized.
// ---------------------------------------------------------------------------
__global__ void init_emb_kernel(const float4* __restrict__ uw,
                                const float4* __restrict__ iw,
                                float4* __restrict__ x0,
                                float4* __restrict__ acc,
                                long nu4, long total4) {
    long i = (long)blockIdx.x * blockDim.x + threadIdx.x;
    if (i >= total4) return;
    float4 v = (i < nu4) ? uw[i] : iw[i - nu4];
    x0[i]  = v;
    acc[i] = v;
}

// ---------------------------------------------------------------------------
// K1: zero the degree array
// ---------------------------------------------------------------------------
__global__ void zero_f32_kernel(float* __restrict__ p, int n) {
    int i = blockIdx.x * blockDim.x + threadIdx.x;
    if (i < n) p[i] = 0.0f;
}

// ---------------------------------------------------------------------------
// K2: degree via float atomic add of 1.0 (integer-valued -> exact, order-free)
// ---------------------------------------------------------------------------
__global__ void deg_kernel(const int* __restrict__ src, float* __restrict__ deg, int nE) {
    int e = blockIdx.x * blockDim.x + threadIdx.x;
    if (e < nE) atomicAdd(&deg[src[e]], 1.0f);
}

// ---------------------------------------------------------------------------
// K3a: per-block exclusive scan of (int)deg; block totals out.  (parallel)
// ---------------------------------------------------------------------------
__global__ __launch_bounds__(SCAN_B) void scan_block_kernel(
        const float* __restrict__ deg, int* __restrict__ rowp,
        int* __restrict__ totals, int n) {
    __shared__ int s[SCAN_B];
    int tid = threadIdx.x;
    int idx = blockIdx.x * SCAN_B + tid;
    int v = (idx < n) ? (int)deg[idx] : 0;
    s[tid] = v;
    __syncthreads();
    for (int off = 1; off < SCAN_B; off <<= 1) {
        int t = (tid >= off) ? s[tid - off] : 0;
        __syncthreads();
        s[tid] += t;
        __syncthreads();
    }
    if (idx < n) rowp[idx] = s[tid] - v;          // block-local exclusive
    if (tid == SCAN_B - 1) totals[blockIdx.x] = s[tid];
}

// ---------------------------------------------------------------------------
// K3b: single small block: exclusive scan of block totals (nb <= 1024 here)
// ---------------------------------------------------------------------------
__global__ __launch_bounds__(SCAN_B) void scan_totals_kernel(int* __restrict__ totals, int nb) {
    __shared__ int s[SCAN_B];
    int tid = threadIdx.x;
    int v = (tid < nb) ? totals[tid] : 0;
    s[tid] = v;
    __syncthreads();
    for (int off = 1; off < SCAN_B; off <<= 1) {
        int t = (tid >= off) ? s[tid - off] : 0;
        __syncthreads();
        s[tid] += t;
        __syncthreads();
    }
    if (tid < nb) totals[tid] = s[tid] - v;       // exclusive
}

// ---------------------------------------------------------------------------
// K3c: add block offsets; duplicate into cursor; rowp[n] = nE
// ---------------------------------------------------------------------------
__global__ void scan_add_kernel(int* __restrict__ rowp, int* __restrict__ cur,
                                const int* __restrict__ totals, int n, int nE) {
    int i = blockIdx.x * blockDim.x + threadIdx.x;
    if (i < n) {
        int r = rowp[i] + totals[i >> 10];
        rowp[i] = r;
        cur[i]  = r;
    }
    if (i == 0) rowp[n] = nE;
}

// ---------------------------------------------------------------------------
// K4: dinv = deg > 0 ? deg^-1/2 : 0
// ---------------------------------------------------------------------------
__global__ void dinv_kernel(const float* __restrict__ deg, float* __restrict__ dinv, int n) {
    int i = blockIdx.x * blockDim.x + threadIdx.x;
    if (i < n) {
        float d = deg[i];
        dinv[i] = (d > 0.0f) ? (1.0f / sqrtf(d)) : 0.0f;
    }
}

// ---------------------------------------------------------------------------
// K5: CSR scatter with fused edge-value computation
// ---------------------------------------------------------------------------
__global__ void scatter_kernel(const int* __restrict__ src, const int* __restrict__ dst,
                               const float* __restrict__ dinv,
                               int* __restrict__ cur,
                               int* __restrict__ cdst, float* __restrict__ cval, int nE) {
    int e = blockIdx.x * blockDim.x + threadIdx.x;
    if (e >= nE) return;
    int s = src[e];
    int d = dst[e];
    int pos = atomicAdd(&cur[s], 1);
    cdst[pos] = d;
    cval[pos] = dinv[s] * dinv[d];
}

// ---------------------------------------------------------------------------
// K6: SpMM, one wave32 per node. Scalarized CSR walk (s_loads). Neighbor rows
// (512B each) are gathered with GLOBAL_LOAD_ASYNC_TO_LDS_B128 into a 4-deep
// per-wave LDS ring: up to 3 gathers in flight (ASYNCcnt) behind the buffer
// being consumed via ds_load_b128 + FMA. Async loads complete in order, so
// s_wait_asynccnt <= 3 (tail: <=2/<=1/<=0) guarantees buffer k is resident.
// Fuses acc += x_new; last layer fuses the *0.25 scale, skips dead x_new store.
// ---------------------------------------------------------------------------
__global__ __launch_bounds__(SPMM_BLOCK) void spmm_kernel(
        const int* __restrict__ rowp,
        const int* __restrict__ cdst,
        const float* __restrict__ cval,
        const float* __restrict__ x,
        float* __restrict__ xn,
        float* __restrict__ acc, int n_nodes, int last, float scale) {
    int lane = threadIdx.x & 31;
    int flat = (int)(((long)blockIdx.x * blockDim.x + threadIdx.x) >> 5);
    int node = __builtin_amdgcn_readfirstlane(flat);   // force SGPR: wave-uniform
    if (node >= n_nodes) return;

    int beg = rowp[node];
    int end = rowp[node + 1];

    float ax = 0.0f, ay = 0.0f, az = 0.0f, aw = 0.0f;

#if HAVE_ASYNC_LDS
    __shared__ alignas(16) float smem[WAVES_PER_BLOCK * PF * D];
    int wib = threadIdx.x >> 5;
    float* bufs = &smem[wib * PF * D];

    // prologue: issue up to PF-1 gathers ahead
    for (int j = beg; j < beg + (PF - 1) && j < end; ++j) {
        const float* g = x + (long)cdst[j] * D + lane * 4;
        __builtin_amdgcn_global_load_async_to_lds_b128(
            (gvec_p)g, (lvec_p)(bufs + ((j - beg) & (PF - 1)) * D + lane * 4), 0, 0);
    }
    asm volatile("" ::: "memory");

    for (int k = beg; k < end; ++k) {
        float v = cval[k];
        float* rd = bufs + ((k - beg) & (PF - 1)) * D;
        if (k + (PF - 1) < end) {
            int j = k + (PF - 1);
            const float* g = x + (long)cdst[j] * D + lane * 4;
            __builtin_amdgcn_global_load_async_to_lds_b128(
                (gvec_p)g, (lvec_p)(bufs + ((j - beg) & (PF - 1)) * D + lane * 4), 0, 0);
        }
        int rem = end - 1 - k;   // loads beyond k that may still be outstanding
        if (rem >= 3)      { WAIT_ASYNC(3); }
        else if (rem == 2) { WAIT_ASYNC(2); }
        else if (rem == 1) { WAIT_ASYNC(1); }
        else               { WAIT_ASYNC(0); }
        asm volatile("" ::: "memory");
        const float4 xv = *(const float4*)(rd + lane * 4);   // ds_load_b128
        ax = fmaf(v, xv.x, ax);
        ay = fmaf(v, xv.y, ay);
        az = fmaf(v, xv.z, az);
        aw = fmaf(v, xv.w, aw);
    }
#else
    for (int k = beg; k < end; ++k) {
        int   d = cdst[k];
        float v = cval[k];
        if (k + 1 < end) {
            __builtin_prefetch(x + (long)cdst[k + 1] * D + lane * 4, 0, 0);
        }
        const float4 xv = *((const float4*)(x + (long)d * D) + lane);
        ax = fmaf(v, xv.x, ax);
        ay = fmaf(v, xv.y, ay);
        az = fmaf(v, xv.z, az);
        aw = fmaf(v, xv.w, aw);
    }
#endif

    long o = (long)node * D + lane * 4;
    if (!last) {
        float4 r; r.x = ax; r.y = ay; r.z = az; r.w = aw;
        *(float4*)(xn + o) = r;
    }
    float4 c = *(const float4*)(acc + o);
    c.x = (c.x + ax) * scale;
    c.y = (c.y + ay) * scale;
    c.z = (c.z + az) * scale;
    c.w = (c.w + aw) * scale;
    *(float4*)(acc + o) = c;
}

// ---------------------------------------------------------------------------
extern "C" void kernel_launch(void* const* d_in, const int* in_sizes, int n_in,
                              void* d_out, int out_size, void* d_ws, size_t ws_size,
                              hipStream_t stream) {
    (void)n_in; (void)out_size; (void)ws_size;

    const float* user_w = (const float*)d_in[0];
    const float* item_w = (const float*)d_in[1];
    const int*   ei     = (const int*)d_in[2];

    const int num_users = in_sizes[0] / D;
    const int num_items = in_sizes[1] / D;
    const int n_nodes   = num_users + num_items;
    const int nE        = in_sizes[2] / 2;   // edge_index is [2, nE] row-major

    const int* src = ei;        // row 0
    const int* dst = ei + nE;   // row 1

    const int nb_scan = (n_nodes + SCAN_B - 1) / SCAN_B;   // 147 here (<=1024 req'd)

    auto ALN = [](size_t x) { return (x + 255) & ~(size_t)255; };
    char* w = (char*)d_ws;
    float* deg  = (float*)w;  w += ALN((size_t)n_nodes * 4);
    float* dinv = (float*)w;  w += ALN((size_t)n_nodes * 4);
    int*   rowp = (int*)w;    w += ALN((size_t)(n_nodes + 1) * 4);
    int*   cur  = (int*)w;    w += ALN((size_t)n_nodes * 4);
    int*   tot  = (int*)w;    w += ALN((size_t)nb_scan * 4);
    int*   cdst = (int*)w;    w += ALN((size_t)nE * 4);
    float* cval = (float*)w;  w += ALN((size_t)nE * 4);
    float* x0   = (float*)w;  w += ALN((size_t)n_nodes * D * 4);
    float* x1   = (float*)w;  w += ALN((size_t)n_nodes * D * 4);
    float* acc  = (float*)d_out;

    const long total4 = (long)n_nodes * D4;
    const long nu4    = (long)num_users * D4;
    const int  B      = 256;

    init_emb_kernel<<<(unsigned)((total4 + B - 1) / B), B, 0, stream>>>(
        (const float4*)user_w, (const float4*)item_w, (float4*)x0, (float4*)acc,
        nu4, total4);

    zero_f32_kernel<<<(n_nodes + B - 1) / B, B, 0, stream>>>(deg, n_nodes);
    deg_kernel<<<(nE + B - 1) / B, B, 0, stream>>>(src, deg, nE);

    // parallel 3-phase exclusive scan of degrees -> rowp (+ cursor copy)
    scan_block_kernel<<<nb_scan, SCAN_B, 0, stream>>>(deg, rowp, tot, n_nodes);
    scan_totals_kernel<<<1, SCAN_B, 0, stream>>>(tot, nb_scan);
    scan_add_kernel<<<(n_nodes + B - 1) / B, B, 0, stream>>>(rowp, cur, tot, n_nodes, nE);

    dinv_kernel<<<(n_nodes + B - 1) / B, B, 0, stream>>>(deg, dinv, n_nodes);
    scatter_kernel<<<(nE + B - 1) / B, B, 0, stream>>>(src, dst, dinv, cur, cdst, cval, nE);

    float* xin = x0;
    float* xout = x1;
    const long spmm_threads = (long)n_nodes * 32;
    for (int l = 0; l < NUM_LAYERS; ++l) {
        const int  last  = (l == NUM_LAYERS - 1) ? 1 : 0;
        const float scl  = last ? (1.0f / (float)(NUM_LAYERS + 1)) : 1.0f;
        spmm_kernel<<<(unsigned)((spmm_threads + SPMM_BLOCK - 1) / SPMM_BLOCK),
                      SPMM_BLOCK, 0, stream>>>(rowp, cdst, cval, xin, xout, acc,
                                               n_nodes, last, scl);
        float* t = xin; xin = xout; xout = t;
    }
}